// GNN_75840532513057
// MI455X (gfx1250) — compile-verified
//
#include <hip/hip_runtime.h>
#include <hip/hip_bf16.h>

// ---------------------------------------------------------------------------
// GIN forward on MI455X (gfx1250, wave32).
//
// Cost model: GEMMs ~5 GFLOP total (negligible); message scatter/gather is
// ~0.5 GB/layer but h/agg (12.8 MB each) are L2-resident (192 MB L2), so the
// kernel is L2-latency/atomic-throughput bound. HBM streams edge_index (8MB)
// + edge_attr (20MB) per layer -> ~100-200us total at 23.3 TB/s.
// MLP uses fp32 WMMA (V_WMMA_F32_16X16X4_F32) to match the fp32 reference
// exactly in precision class while exercising the matrix pipe.
// ---------------------------------------------------------------------------

#define N_NODES 50000
#define N_EDGES 1000000
#define EMB     64
#define HID     128
#define LAYERS  3
#define XF      7
#define EF      5
#define BGRAPH  250
#define BN_EPS  1e-5f

typedef float v2f __attribute__((ext_vector_type(2)));
typedef float v8f __attribute__((ext_vector_type(8)));

// ---------------------------------------------------------------------------
// h = x @ xW + xb            [N,7]x[7,64] -> [N,64]
// ---------------------------------------------------------------------------
__global__ void __launch_bounds__(256) k_embed(const float* __restrict__ x,
                                               const float* __restrict__ xW,
                                               const float* __restrict__ xb,
                                               float* __restrict__ h) {
  int t = blockIdx.x * 256 + threadIdx.x;
  if (t >= N_NODES * EMB) return;
  int v = t >> 6, k = t & 63;
  float acc = xb[k];
#pragma unroll
  for (int f = 0; f < XF; ++f) acc += x[v * XF + f] * xW[f * EMB + k];
  h[t] = acc;
}

// ---------------------------------------------------------------------------
// agg[v] = h[v] + (8*eW[l][0,:] + eb[l])   (self-loop contribution)
// ---------------------------------------------------------------------------
__global__ void __launch_bounds__(256) k_init_agg(const float* __restrict__ h,
                                                  const float* __restrict__ eWl,
                                                  const float* __restrict__ ebl,
                                                  float* __restrict__ agg) {
  int t = blockIdx.x * 256 + threadIdx.x;
  if (t >= N_NODES * EMB) return;
  int k = t & 63;
  agg[t] = h[t] + 8.0f * eWl[k] + ebl[k];
}

// ---------------------------------------------------------------------------
// Edge scatter: one wave per edge. Edge index/attr loads are wave-uniform
// (forced via readfirstlane -> SMEM path); h-row gather and agg atomics are
// 64 coalesced fp32 lanes (2 per lane). agg lives in L2.
// ---------------------------------------------------------------------------
__global__ void __launch_bounds__(256) k_edge_scatter(
    const float* __restrict__ h, const float* __restrict__ edge_attr,
    const int* __restrict__ edge_index, const float* __restrict__ eWl,
    const float* __restrict__ ebl, float* __restrict__ agg) {
  __shared__ float sW[EF * EMB + EMB];
  for (int i = threadIdx.x; i < EF * EMB + EMB; i += 256)
    sW[i] = (i < EF * EMB) ? eWl[i] : ebl[i - EF * EMB];
  __syncthreads();

  int wv = __builtin_amdgcn_readfirstlane((int)(threadIdx.x >> 5));
  int e = blockIdx.x * 8 + wv;
  if (e >= N_EDGES) return;
  int lane = threadIdx.x & 31;

  int src = edge_index[e];              // scalar load (wave-uniform)
  int dst = edge_index[N_EDGES + e];    // scalar load

  // Speculative prefetch of the edge stream + a future source row.
  __builtin_prefetch(edge_attr + (size_t)(e + 512) * EF, 0, 0);
  int srcf = edge_index[(e + 64 < N_EDGES) ? (e + 64) : e];
  __builtin_prefetch(h + (size_t)srcf * EMB, 0, 0);

  float a0 = edge_attr[e * EF + 0];     // scalar loads
  float a1 = edge_attr[e * EF + 1];
  float a2 = edge_attr[e * EF + 2];
  float a3 = edge_attr[e * EF + 3];
  float a4 = edge_attr[e * EF + 4];

  int c0 = lane, c1 = lane + 32;
  float e0 = sW[EF * EMB + c0] + a0 * sW[0 * EMB + c0] + a1 * sW[1 * EMB + c0] +
             a2 * sW[2 * EMB + c0] + a3 * sW[3 * EMB + c0] + a4 * sW[4 * EMB + c0];
  float e1 = sW[EF * EMB + c1] + a0 * sW[0 * EMB + c1] + a1 * sW[1 * EMB + c1] +
             a2 * sW[2 * EMB + c1] + a3 * sW[3 * EMB + c1] + a4 * sW[4 * EMB + c1];

  const float* hr = h + (size_t)src * EMB;
  float* ar = agg + (size_t)dst * EMB;
  atomicAdd(&ar[c0], hr[c0] + e0);
  atomicAdd(&ar[c1], hr[c1] + e1);
}

// ---------------------------------------------------------------------------
// h2 = relu(agg @ W1 + b1) @ W2 + b2   via V_WMMA_F32_16X16X4_F32.
// One wave owns a 16-row tile. GEMM1 (K=64, N=128): 8 n-tiles x 16 k-steps.
// ReLU intermediate re-striped through LDS (C layout: lane=N -> A layout:
// lane=M). GEMM2 (K=128, N=64): 4 n-tiles x 32 k-steps. 256 WMMAs per tile.
// Fragment layouts per ISA 7.12.2:
//   A 16x4 f32: M = lane%16; VGPR j holds K = 4*step + 2*(lane/16) + j
//   B 4x16 f32: N = lane%16; VGPR j holds K = 4*step + 2*(lane/16) + j
//   C 16x16:    N = lane%16; VGPR i holds M = i + 8*(lane/16)
// ---------------------------------------------------------------------------
#define IST 132  // padded LDS stride for the 16x128 intermediate

__global__ void __launch_bounds__(128) k_mlp(const float* __restrict__ agg,
                                             const float* __restrict__ W1,
                                             const float* __restrict__ b1,
                                             const float* __restrict__ W2,
                                             const float* __restrict__ b2,
                                             float* __restrict__ h2) {
  __shared__ float sI[4 * 16 * IST];  // 33.8 KB: per-wave 16x128 intermediate
  int wave = threadIdx.x >> 5, lane = threadIdx.x & 31;
  int hh = lane >> 4, ln = lane & 15;  // half-select, lane-in-half
  int r0 = (blockIdx.x * 4 + wave) * 16;
  if (r0 >= N_NODES) return;
  float* myI = sI + wave * 16 * IST;

  // A fragments for GEMM1: row (r0+ln), 16 K-steps of 2 contiguous floats.
  const float* arow = agg + (size_t)(r0 + ln) * EMB + 2 * hh;
  v2f afrag[16];
#pragma unroll
  for (int s = 0; s < 16; ++s) afrag[s] = *(const v2f*)(arow + 4 * s);

  // GEMM1 + bias + ReLU -> LDS
#pragma unroll
  for (int nt = 0; nt < 8; ++nt) {
    v8f c = {};
#pragma unroll
    for (int s = 0; s < 16; ++s) {
      int kr = 4 * s + 2 * hh;
      v2f b;
      b.x = W1[kr * HID + nt * 16 + ln];
      b.y = W1[(kr + 1) * HID + nt * 16 + ln];
      c = __builtin_amdgcn_wmma_f32_16x16x4_f32(false, afrag[s], false, b,
                                                (short)0, c, false, false);
    }
    float bias = b1[nt * 16 + ln];
#pragma unroll
    for (int i = 0; i < 8; ++i) {
      float v = c[i] + bias;
      myI[(i + 8 * hh) * IST + nt * 16 + ln] = v > 0.0f ? v : 0.0f;
    }
  }

  // A fragments for GEMM2 from LDS (lane%16 = M row, contiguous K pairs).
  v2f a2[32];
#pragma unroll
  for (int s = 0; s < 32; ++s)
    a2[s] = *(const v2f*)(myI + ln * IST + 4 * s + 2 * hh);

  // GEMM2 + bias -> h2
#pragma unroll
  for (int nt = 0; nt < 4; ++nt) {
    v8f c = {};
#pragma unroll
    for (int s = 0; s < 32; ++s) {
      int kr = 4 * s + 2 * hh;
      v2f b;
      b.x = W2[kr * EMB + nt * 16 + ln];
      b.y = W2[(kr + 1) * EMB + nt * 16 + ln];
      c = __builtin_amdgcn_wmma_f32_16x16x4_f32(false, a2[s], false, b,
                                                (short)0, c, false, false);
    }
    float bias = b2[nt * 16 + ln];
#pragma unroll
    for (int i = 0; i < 8; ++i)
      h2[(size_t)(r0 + i + 8 * hh) * EMB + nt * 16 + ln] = c[i] + bias;
  }
}

// ---------------------------------------------------------------------------
// Batch-norm statistics: per-channel sum / sumsq over all N nodes.
// ---------------------------------------------------------------------------
__global__ void k_zero_stats(float* __restrict__ stats) {
  if (threadIdx.x < 128) stats[threadIdx.x] = 0.0f;
}

__global__ void __launch_bounds__(256) k_bn_stats(const float* __restrict__ h2,
                                                  float* __restrict__ stats) {
  __shared__ float ssum[256], ssq[256];
  int k = threadIdx.x & 63;
  int grp = threadIdx.x >> 6;  // 4 row-groups per block
  float sum = 0.0f, sq = 0.0f;
  for (int v = blockIdx.x * 4 + grp; v < N_NODES; v += gridDim.x * 4) {
    float val = h2[(size_t)v * EMB + k];
    sum += val;
    sq += val * val;
  }
  ssum[threadIdx.x] = sum;
  ssq[threadIdx.x] = sq;
  __syncthreads();
  if (grp == 0) {
    sum = ssum[k] + ssum[64 + k] + ssum[128 + k] + ssum[192 + k];
    sq = ssq[k] + ssq[64 + k] + ssq[128 + k] + ssq[192 + k];
    atomicAdd(&stats[k], sum);
    atomicAdd(&stats[64 + k], sq);
  }
}

// ---------------------------------------------------------------------------
// Apply BN (+ELU except last layer); writes next-layer h.
// ---------------------------------------------------------------------------
__global__ void __launch_bounds__(256) k_bn_apply(
    const float* __restrict__ h2, const float* __restrict__ stats,
    const float* __restrict__ gamma, const float* __restrict__ beta,
    float* __restrict__ h, int do_elu) {
  int t = blockIdx.x * 256 + threadIdx.x;
  if (t >= N_NODES * EMB) return;
  int k = t & 63;
  const float inv_n = 1.0f / (float)N_NODES;
  float mean = stats[k] * inv_n;
  float var = stats[64 + k] * inv_n - mean * mean;
  float rstd = rsqrtf(var + BN_EPS);
  float hn = (h2[t] - mean) * rstd * gamma[k] + beta[k];
  if (do_elu) hn = hn > 0.0f ? hn : (expf(hn) - 1.0f);
  h[t] = hn;
}

// ---------------------------------------------------------------------------
// out[b] = h[last_idx[b]]
// ---------------------------------------------------------------------------
__global__ void __launch_bounds__(256) k_gather(const float* __restrict__ h,
                                                const int* __restrict__ last_idx,
                                                float* __restrict__ out) {
  int t = blockIdx.x * 256 + threadIdx.x;
  if (t >= BGRAPH * EMB) return;
  int b = t >> 6, k = t & 63;
  out[t] = h[(size_t)last_idx[b] * EMB + k];
}

// ---------------------------------------------------------------------------
extern "C" void kernel_launch(void* const* d_in, const int* in_sizes, int n_in,
                              void* d_out, int out_size, void* d_ws,
                              size_t ws_size, hipStream_t stream) {
  const float* x         = (const float*)d_in[0];
  const float* edge_attr = (const float*)d_in[1];
  const int*   edge_index= (const int*)d_in[2];
  const int*   last_idx  = (const int*)d_in[3];
  const float* xW        = (const float*)d_in[4];
  const float* xb        = (const float*)d_in[5];
  const float* eW        = (const float*)d_in[6];
  const float* eb        = (const float*)d_in[7];
  const float* W1        = (const float*)d_in[8];
  const float* b1        = (const float*)d_in[9];
  const float* W2        = (const float*)d_in[10];
  const float* b2        = (const float*)d_in[11];
  const float* gamma     = (const float*)d_in[12];
  const float* beta      = (const float*)d_in[13];

  // Workspace: h, agg, h2 (N*64 floats each) + stats (128 floats) ~= 38.4 MB
  float* h     = (float*)d_ws;
  float* agg   = h + (size_t)N_NODES * EMB;
  float* h2    = agg + (size_t)N_NODES * EMB;
  float* stats = h2 + (size_t)N_NODES * EMB;

  const int ne_grid = (N_NODES * EMB + 255) / 256;  // 12500

  k_embed<<<ne_grid, 256, 0, stream>>>(x, xW, xb, h);
  for (int l = 0; l < LAYERS; ++l) {
    const float* eWl = eW + l * EF * EMB;
    const float* ebl = eb + l * EMB;
    k_init_agg<<<ne_grid, 256, 0, stream>>>(h, eWl, ebl, agg);
    k_edge_scatter<<<(N_EDGES + 7) / 8, 256, 0, stream>>>(h, edge_attr,
                                                          edge_index, eWl, ebl,
                                                          agg);
    k_mlp<<<(N_NODES / 16 + 3) / 4, 128, 0, stream>>>(
        agg, W1 + l * EMB * HID, b1 + l * HID, W2 + l * HID * EMB,
        b2 + l * EMB, h2);
    k_zero_stats<<<1, 128, 0, stream>>>(stats);
    k_bn_stats<<<128, 256, 0, stream>>>(h2, stats);
    k_bn_apply<<<ne_grid, 256, 0, stream>>>(h2, stats, gamma + l * EMB,
                                            beta + l * EMB, h,
                                            (l < LAYERS - 1) ? 1 : 0);
  }
  k_gather<<<(BGRAPH * EMB + 255) / 256, 256, 0, stream>>>(h, last_idx,
                                                           (float*)d_out);
}